// MultiHeadAttention_37538014167348
// MI455X (gfx1250) — compile-verified
//
#include <hip/hip_runtime.h>

// Problem constants from the reference: B=4, S=1024, H=768.
// Algebraic simplification (see round 0): softmax rows sum to 1 for any
// scores/mask, and the reference's 'bhqk,bhvd->bhqd' einsum factorizes, so
//   out[b,s,:] = (sum_s x[b,s,:]) @ Wv + S*bv      (independent of s)
#define B_ 4
#define S_ 1024
#define H_ 768
#define MPAD_ 16  // A matrix padded from 4 rows to 16 for WMMA

typedef float v2f __attribute__((ext_vector_type(2)));
typedef float v8f __attribute__((ext_vector_type(8)));

// ---------------------------------------------------------------------------
// Kernel 1: partial column sums of x over the sequence dimension.
// x is [B, S, H] complex64 (interleaved float2). Each block handles 128
// consecutive h-columns for one (batch, s-chunk); chunk = 64 sequence rows.
// psum layout: [B][16][H] complex.
// ---------------------------------------------------------------------------
__global__ __launch_bounds__(128) void k_partial(const float2* __restrict__ x,
                                                 float2* __restrict__ psum) {
    const int h = blockIdx.y * 128 + threadIdx.x;
    const int b = blockIdx.x;
    const int c = blockIdx.z;  // s-chunk 0..15
    const float2* p = x + ((size_t)(b * S_ + c * 64)) * H_ + h;
    float re = 0.f, im = 0.f;
#pragma unroll 8
    for (int s = 0; s < 64; ++s) {
        float2 v = p[(size_t)s * H_];
        re += v.x;
        im += v.y;
    }
    psum[(size_t)(b * 16 + c) * H_ + h] = make_float2(re, im);
}

// ---------------------------------------------------------------------------
// Kernel 2: fold the 16 partials into a ZERO-PADDED, DEINTERLEAVED A matrix:
//   Are[16][H], Aim[16][H]  (rows 0..3 = batch sums, rows 4..15 = 0)
// Padding + planar layout makes the WMMA kernel's A-fragment loads
// unconditional aligned b64 loads (no exec-mask predication, no shuffles).
// ---------------------------------------------------------------------------
__global__ __launch_bounds__(128) void k_finish(const float2* __restrict__ psum,
                                                float* __restrict__ Are,
                                                float* __restrict__ Aim) {
    const int idx = blockIdx.x * 128 + threadIdx.x;  // 0 .. MPAD_*H_-1
    const int m = idx / H_;
    const int h = idx % H_;
    float re = 0.f, im = 0.f;
    if (m < B_) {
#pragma unroll
        for (int c = 0; c < 16; ++c) {
            float2 v = psum[(size_t)(m * 16 + c) * H_ + h];
            re += v.x;
            im += v.y;
        }
    }
    Are[idx] = re;
    Aim[idx] = im;
}

// ---------------------------------------------------------------------------
// Kernel 3: rows[b][h] = A[b][:] @ Wv[:][h] + S*bv[h]   (complex GEMM)
// via V_WMMA_F32_16X16X4_F32. One wave per 16-wide N tile (48 blocks).
//   A frag  (16x4 f32, 2 VGPR): M = lane&15, K = k0 + 2*(lane>>4) + vgpr
//   B frag  (4x16 f32, 2 VGPR): N = lane&15, K = k0 + 2*(lane>>4) + vgpr
//   C/D     (16x16 f32, 8 VGPR): lanes 0-15 vgpr r -> (M=r, N=lane)
// Complex product = 4 real WMMAs per K-step:
//   C_re += A_re*B_re + (-A_im)*B_im ; C_im += A_re*B_im + A_im*B_re
// (F32 WMMA NEG bits only allow C-negate, so A_im is sign-flipped in VALU.)
// ---------------------------------------------------------------------------
__global__ __launch_bounds__(32) void k_gemm_wmma(const float* __restrict__ Are,
                                                  const float* __restrict__ Aim,
                                                  const float2* __restrict__ Wv,
                                                  const float2* __restrict__ bv,
                                                  float2* __restrict__ rows) {
    const int lane  = threadIdx.x;
    const int mrow  = lane & 15;
    const int khalf = lane >> 4;  // 0 or 1
    const int ncol  = blockIdx.x * 16 + (lane & 15);

    const float* pre = Are + mrow * H_ + 2 * khalf;  // even float idx: 8B aligned
    const float* pim = Aim + mrow * H_ + 2 * khalf;

    v8f acc_re = {};
    v8f acc_im = {};

    for (int k0 = 0; k0 < H_; k0 += 4) {
        // Unconditional planar A loads (rows 4..15 pre-zeroed).
        v2f a_re = *(const v2f*)(pre + k0);
        v2f a_im = *(const v2f*)(pim + k0);
        v2f a_imn;
        a_imn[0] = -a_im[0];
        a_imn[1] = -a_im[1];

        const int ka = k0 + 2 * khalf;
        float2 b0 = Wv[(size_t)ka * H_ + ncol];
        float2 b1 = Wv[(size_t)(ka + 1) * H_ + ncol];
        v2f b_re, b_im;
        b_re[0] = b0.x; b_re[1] = b1.x;
        b_im[0] = b0.y; b_im[1] = b1.y;

        acc_re = __builtin_amdgcn_wmma_f32_16x16x4_f32(false, a_re,  false, b_re,
                                                       (short)0, acc_re, false, false);
        acc_re = __builtin_amdgcn_wmma_f32_16x16x4_f32(false, a_imn, false, b_im,
                                                       (short)0, acc_re, false, false);
        acc_im = __builtin_amdgcn_wmma_f32_16x16x4_f32(false, a_re,  false, b_im,
                                                       (short)0, acc_im, false, false);
        acc_im = __builtin_amdgcn_wmma_f32_16x16x4_f32(false, a_im,  false, b_re,
                                                       (short)0, acc_im, false, false);
    }

    // Rows M=0..3 (= batches) live in VGPRs 0..3 of lanes 0..15.
    if (lane < 16) {
        float2 bias = bv[ncol];
#pragma unroll
        for (int r = 0; r < B_; ++r) {
            rows[r * H_ + ncol] =
                make_float2(acc_re[r] + (float)S_ * bias.x,
                            acc_im[r] + (float)S_ * bias.y);
        }
    }
}

// ---------------------------------------------------------------------------
// Kernel 4: broadcast rows[b][:] to every sequence position.
// out[b][s][h] = rows[b][h]. float4 (B128) stores; the 24 KB rows table
// stays cache-resident. This is the dominant 25 MB of HBM traffic
// (50 MB total with the x read -> ~2.2 us at 23.3 TB/s).
// ---------------------------------------------------------------------------
__global__ __launch_bounds__(256) void k_broadcast(const float2* __restrict__ rows,
                                                   float4* __restrict__ out) {
    const size_t i  = (size_t)blockIdx.x * 256 + threadIdx.x;  // of B*S*H/2
    const size_t c0 = i * 2;                                   // complex index
    const int h = (int)(c0 % H_);
    const int b = (int)(c0 / ((size_t)S_ * H_));
    float2 r0 = rows[b * H_ + h];
    float2 r1 = rows[b * H_ + h + 1];
    out[i] = make_float4(r0.x, r0.y, r1.x, r1.y);
}

// ---------------------------------------------------------------------------
// Inputs (setup_inputs order): 0=x, 1=Wq, 2=bq, 3=Wk, 4=bk, 5=Wv, 6=bv, 7=mask.
// Only x, Wv, bv matter after the softmax-row-sum identity.
// ---------------------------------------------------------------------------
extern "C" void kernel_launch(void* const* d_in, const int* in_sizes, int n_in,
                              void* d_out, int out_size, void* d_ws, size_t ws_size,
                              hipStream_t stream) {
    (void)in_sizes; (void)n_in; (void)out_size; (void)ws_size;

    const float2* x  = (const float2*)d_in[0];
    const float2* Wv = (const float2*)d_in[5];
    const float2* bv = (const float2*)d_in[6];

    // Scratch layout, ~0.6 MB total, fully rewritten every call:
    float2* psum = (float2*)d_ws;                        // B*16*H complex
    float*  Are  = (float*)(psum + (size_t)B_ * 16 * H_); // MPAD*H floats
    float*  Aim  = Are + (size_t)MPAD_ * H_;              // MPAD*H floats
    float2* rows = (float2*)(Aim + (size_t)MPAD_ * H_);   // B*H complex

    k_partial<<<dim3(B_, H_ / 128, 16), 128, 0, stream>>>(x, psum);
    k_finish<<<dim3((MPAD_ * H_) / 128), 128, 0, stream>>>(psum, Are, Aim);
    k_gemm_wmma<<<dim3(H_ / 16), 32, 0, stream>>>(Are, Aim, Wv, bv, rows);
    k_broadcast<<<dim3((unsigned)((size_t)B_ * S_ * H_ / 2 / 256)), 256, 0, stream>>>(
        rows, (float4*)d_out);
}